// CGGCN_5446018531350
// MI455X (gfx1250) — compile-verified
//
#include <hip/hip_runtime.h>
#include <hip/hip_bf16.h>
#include <stdint.h>

typedef float v2f __attribute__((ext_vector_type(2)));
typedef float v8f __attribute__((ext_vector_type(8)));

#define BD   64
#define ND   2048
#define DD   256
#define TD   131072        // B*N
#define ED   1048576       // edges
#define KCH  32            // K-chunk rows staged per async pipeline stage

// ---------------------------------------------------------------------------
// GEMM: {m0,m1,h} = feat @ {W0, W1, Wself}   (fp32 WMMA 16x16x4)
// A tile (16x256) staged via async-to-LDS cooperatively; B chunks (32x16)
// double-buffered per wave via async-to-LDS with ASYNCcnt pipelining.
// ---------------------------------------------------------------------------
__global__ __launch_bounds__(256) void cggcn_gemm(
    const float* __restrict__ feat, const float* __restrict__ W2,
    const float* __restrict__ Wself, float* __restrict__ m0,
    float* __restrict__ m1, float* __restrict__ h)
{
    __shared__ __align__(16) float As[16 * 256];        // 16 KB A tile
    __shared__ __align__(16) float Bs[8][2][KCH * 16];  // 8 waves x dbl x 2 KB

    const int t      = threadIdx.x;
    const int wave   = t >> 5;
    const int lane   = t & 31;
    const int half   = lane >> 4;          // 0: lanes 0-15, 1: lanes 16-31
    const int lane16 = lane & 15;
    const int rb     = blockIdx.x;         // 16-row block of feat

    // ---- async-stage contiguous 16x256 A tile (4096 floats, 4x b128/thread)
    {
        const char*    gbase = (const char*)(feat + (size_t)rb * 16 * 256);
        const uint32_t lbase = (uint32_t)(uintptr_t)(void*)As;
#pragma unroll
        for (int i = 0; i < 4; ++i) {
            const uint32_t off = (uint32_t)(t + 256 * i) * 16u;
            asm volatile("global_load_async_to_lds_b128 %0, %1, %2"
                         :: "v"(lbase + off), "v"(off), "s"(gbase) : "memory");
        }
    }
    asm volatile("s_wait_asynccnt 0x0" ::: "memory");
    __syncthreads();

    for (int task = wave; task < 48; task += 8) {
        const int w  = task >> 4;          // 0 -> W0, 1 -> W1, 2 -> Wself
        const int ct = task & 15;          // column tile (16 cols)
        const float* Wp  = (w == 2) ? Wself : (W2 + (size_t)w * DD * DD);
        float*       Out = (w == 0) ? m0 : ((w == 1) ? m1 : h);

        // per-lane addressing for B chunk staging (4x b128 per lane per chunk)
        const char* gb = (const char*)(Wp + ct * 16);

        auto issue_chunk = [&](int kc, int buf) {
            const uint32_t lb = (uint32_t)(uintptr_t)(void*)&Bs[wave][buf][0];
#pragma unroll
            for (int i = 0; i < 4; ++i) {
                const int f   = lane + 32 * i;   // float4 id 0..127
                const int row = f >> 2;          // 0..31 within chunk
                const int cg  = (f & 3) * 4;     // col group
                const uint32_t goff = (uint32_t)(((kc * KCH + row) * DD + cg) * 4);
                const uint32_t loff = (uint32_t)((row * 16 + cg) * 4);
                asm volatile("global_load_async_to_lds_b128 %0, %1, %2"
                             :: "v"(lb + loff), "v"(goff), "s"(gb) : "memory");
            }
        };

        v8f c = {};                        // fp32 accumulator (8 VGPRs)
        issue_chunk(0, 0);

        for (int kc = 0; kc < 8; ++kc) {   // 8 chunks of K=32
            if (kc < 7) {
                issue_chunk(kc + 1, (kc + 1) & 1);
                // in-order async completion: <=4 outstanding => chunk kc ready
                asm volatile("s_wait_asynccnt 0x4" ::: "memory");
            } else {
                asm volatile("s_wait_asynccnt 0x0" ::: "memory");
            }
            const float* bbuf  = &Bs[wave][kc & 1][0];
            const int    kbase = kc * KCH;
#pragma unroll
            for (int kk = 0; kk < KCH; kk += 4) {
                // A 16x4: lanes 0-15 hold K=kk,kk+1 ; lanes 16-31 hold K=kk+2,kk+3
                v2f a = *(const v2f*)(As + lane16 * 256 + kbase + kk + 2 * half);
                // B 4x16: VGPR0 = rows kk / kk+2, VGPR1 = rows kk+1 / kk+3
                v2f b;
                b.x = bbuf[(kk + 2 * half) * 16 + lane16];
                b.y = bbuf[(kk + 2 * half + 1) * 16 + lane16];
                c = __builtin_amdgcn_wmma_f32_16x16x4_f32(
                        false, a, false, b, (short)0, c, false, false);
            }
        }
        // C/D 16x16: VGPR v -> row (v + 8*half), col lane16
        float* Op = Out + ((size_t)rb * 16) * DD + ct * 16;
#pragma unroll
        for (int v = 0; v < 8; ++v)
            Op[(size_t)(v + 8 * half) * DD + lane16] = c[v];
    }
}

// ---------------------------------------------------------------------------
// Edge aggregation: h[dst] += (etype ? m1 : m0)[src]   (wave per edge)
// ---------------------------------------------------------------------------
__global__ __launch_bounds__(256) void cggcn_edges(
    const int* __restrict__ src, const int* __restrict__ dst,
    const int* __restrict__ et, const float* __restrict__ m0,
    const float* __restrict__ m1, float* __restrict__ h)
{
    const int wid  = blockIdx.x * 8 + (threadIdx.x >> 5);
    const int lane = threadIdx.x & 31;
    if (wid >= ED) return;
    const int s = src[wid], d = dst[wid];
    const float4* row4 = (const float4*)((et[wid] ? m1 : m0) + (size_t)s * DD);
    float*        out  = h + (size_t)d * DD;
    const float4 a = row4[lane];        // cols 4*lane .. 4*lane+3
    const float4 b = row4[lane + 32];   // cols 128+4*lane ..
    atomicAdd(out + 4 * lane + 0,   a.x);
    atomicAdd(out + 4 * lane + 1,   a.y);
    atomicAdd(out + 4 * lane + 2,   a.z);
    atomicAdd(out + 4 * lane + 3,   a.w);
    atomicAdd(out + 4 * lane + 128, b.x);
    atomicAdd(out + 4 * lane + 129, b.y);
    atomicAdd(out + 4 * lane + 130, b.z);
    atomicAdd(out + 4 * lane + 131, b.w);
}

// ---------------------------------------------------------------------------
// new_rel_emds = batch_rel_emds, then scatter rows r=index2+1 with base + h
// ---------------------------------------------------------------------------
__global__ __launch_bounds__(256) void cggcn_relcopy(
    const float* __restrict__ brel, float* __restrict__ outrel)
{
    const size_t i = (size_t)blockIdx.x * 256 + threadIdx.x;   // < B*D*D
    outrel[i] = brel[i];
}

__global__ __launch_bounds__(256) void cggcn_relscat(
    const float* __restrict__ brel, const float* __restrict__ h,
    const int* __restrict__ index2, float* __restrict__ outrel)
{
    const int wid  = blockIdx.x * 8 + (threadIdx.x >> 5);      // node id < T
    const int lane = threadIdx.x & 31;
    if (wid >= TD) return;
    const int r = index2[wid] + 1;
    if (r <= 0 || r >= DD) return;                             // masked / OOB
    const int b = wid >> 11;                                   // / N
    const float* hrow = h    + (size_t)wid * DD;
    const float* base = brel + ((size_t)b * DD + r) * DD;
    float*       orow = outrel + ((size_t)b * DD + r) * DD;
#pragma unroll
    for (int j = 0; j < 8; ++j) {
        const int dcol = lane + 32 * j;
        orow[dcol] = base[dcol] + hrow[dcol];                  // .set semantics
    }
}

// ---------------------------------------------------------------------------
// target_rel_emd[b] = new_rel_emds[b, rel_label[b*N]+1]
// ---------------------------------------------------------------------------
__global__ __launch_bounds__(256) void cggcn_target(
    const float* __restrict__ outrel, const int* __restrict__ rel_label,
    float* __restrict__ tgt)
{
    const int b = blockIdx.x, t = threadIdx.x;
    const int tar = rel_label[b * ND] + 1;
    tgt[b * DD + t] = outrel[((size_t)b * DD + tar) * DD + t];
}

// ---------------------------------------------------------------------------
// path_agg: per-graph alpha = h . target, masked weighted mean over nodes
// ---------------------------------------------------------------------------
__global__ __launch_bounds__(256) void cggcn_path(
    const float* __restrict__ h, const float* __restrict__ tgt,
    const int* __restrict__ index1, float* __restrict__ outp)
{
    __shared__ float st[256];
    __shared__ float sacc[256];
    __shared__ float scnt;
    const int b = blockIdx.x, t = threadIdx.x;
    const int wave = t >> 5, lane = t & 31;

    st[t]   = tgt[b * DD + t];
    sacc[t] = 0.0f;
    if (t == 0) scnt = 0.0f;
    __syncthreads();

    float acc[8];
#pragma unroll
    for (int j = 0; j < 8; ++j) acc[j] = 0.0f;
    float cnt = 0.0f;

    const float* hb = h + (size_t)b * ND * DD;
    const int*   i1 = index1 + b * ND;

    for (int n = wave; n < ND; n += 8) {
        float v[8];
        float dp = 0.0f;
#pragma unroll
        for (int j = 0; j < 8; ++j) {
            v[j] = hb[(size_t)n * DD + lane + 32 * j];
            dp  += v[j] * st[lane + 32 * j];
        }
#pragma unroll
        for (int off = 16; off > 0; off >>= 1)
            dp += __shfl_xor(dp, off, 32);
        const float m = (i1[n] != 0) ? 1.0f : 0.0f;
        const float a = dp * m;
#pragma unroll
        for (int j = 0; j < 8; ++j) acc[j] += a * v[j];
        cnt += m;
    }
#pragma unroll
    for (int j = 0; j < 8; ++j)
        atomicAdd(&sacc[lane + 32 * j], acc[j]);               // ds_add_f32
    if (lane == 0) atomicAdd(&scnt, cnt);
    __syncthreads();

    const float c = scnt;
    outp[b * DD + t] = (c > 0.0f) ? sacc[t] / fmaxf(c, 1.0f) : 0.0f;
}

// ---------------------------------------------------------------------------
// node_repr = relu(h) in place (float4 grid)
// ---------------------------------------------------------------------------
__global__ __launch_bounds__(256) void cggcn_relu(float* __restrict__ h)
{
    const size_t i = (size_t)blockIdx.x * 256 + threadIdx.x;   // float4 index
    float4* p = (float4*)h;
    float4 v = p[i];
    v.x = fmaxf(v.x, 0.0f); v.y = fmaxf(v.y, 0.0f);
    v.z = fmaxf(v.z, 0.0f); v.w = fmaxf(v.w, 0.0f);
    p[i] = v;
}

extern "C" void kernel_launch(void* const* d_in, const int* in_sizes, int n_in,
                              void* d_out, int out_size, void* d_ws, size_t ws_size,
                              hipStream_t stream)
{
    const float* feat      = (const float*)d_in[0];
    const float* brel      = (const float*)d_in[1];
    const float* W2        = (const float*)d_in[2];   // [2, D, D]
    const float* Wself     = (const float*)d_in[3];
    const int*   src       = (const int*)d_in[4];
    const int*   dst       = (const int*)d_in[5];
    const int*   etype     = (const int*)d_in[6];
    const int*   index1    = (const int*)d_in[7];
    const int*   index2    = (const int*)d_in[8];
    const int*   rel_label = (const int*)d_in[9];

    float* out    = (float*)d_out;
    float* outrel = out;                               // B*D*D = 4194304
    float* tgt    = out + (size_t)BD * DD * DD;        // +16384
    float* outp   = tgt + (size_t)BD * DD;             // +16384
    float* h      = outp + (size_t)BD * DD;            // T*D (node_repr slot)

    float* m0 = (float*)d_ws;                          // T*D
    float* m1 = m0 + (size_t)TD * DD;                  // T*D

    cggcn_gemm   <<<TD / 16,        256, 0, stream>>>(feat, W2, Wself, m0, m1, h);
    cggcn_edges  <<<ED / 8,         256, 0, stream>>>(src, dst, etype, m0, m1, h);
    cggcn_relcopy<<<BD * DD,        256, 0, stream>>>(brel, outrel);
    cggcn_relscat<<<TD / 8,         256, 0, stream>>>(brel, h, index2, outrel);
    cggcn_target <<<BD,             256, 0, stream>>>(outrel, rel_label, tgt);
    cggcn_path   <<<BD,             256, 0, stream>>>(h, tgt, index1, outp);
    cggcn_relu   <<<(TD * DD) / 1024, 256, 0, stream>>>(h);
}